// QBatchNorm_60344290508886
// MI455X (gfx1250) — compile-verified
//
#include <hip/hip_runtime.h>

typedef float v2f __attribute__((ext_vector_type(2)));
typedef float v4f __attribute__((ext_vector_type(4)));
typedef float v8f __attribute__((ext_vector_type(8)));

#define C_DIM   128
#define N_DIM   64
#define HW      3136                   // 56*56
#define PLANES  (C_DIM * N_DIM)        // 8192
#define NUMEL_PER_CH (N_DIM * HW)      // 200704
#define BN_EPS  1e-5f

static __device__ __forceinline__ v8f wmma_rowsum(v2f a, v2f ones, v8f c) {
    // D = A(16x4 f32) * ones(4x16) + C : accumulates row-sums of A into C.
    return __builtin_amdgcn_wmma_f32_16x16x4_f32(
        false, a, false, ones, (short)0, c, false, false);
}

// ---------------------------------------------------------------------------
// Pass 1: per-(n,c) plane partial sum & sum-of-squares through the f32 WMMA
// pipe. 784 float4 per plane = 3 full 256-thread b128 iterations (768) plus a
// 16-float4 tail that wave 0 consumes as 32 float2s, so EXEC is all-1s at
// every WMMA. Four independent accumulators keep the matrix pipe pipelined.
// ---------------------------------------------------------------------------
__global__ __launch_bounds__(256) void bn_partials(const float* __restrict__ x,
                                                   float* __restrict__ psum,
                                                   float* __restrict__ psq) {
    const int c = blockIdx.x;
    const int n = blockIdx.y;
    const int plane = n * C_DIM + c;
    const v4f* __restrict__ xp4 = (const v4f*)(x + (long)plane * HW);
    const int tid = threadIdx.x;

    v2f ones; ones.x = 1.0f; ones.y = 1.0f;
    v8f s0 = {}, s1 = {}, q0 = {}, q1 = {};

    #pragma unroll
    for (int k = 0; k < 3; ++k) {
        v4f v = xp4[k * 256 + tid];                     // global_load_b128
        v2f a0; a0.x = v.x;       a0.y = v.y;
        v2f a1; a1.x = v.z;       a1.y = v.w;
        v2f b0; b0.x = v.x * v.x; b0.y = v.y * v.y;
        v2f b1; b1.x = v.z * v.z; b1.y = v.w * v.w;
        s0 = wmma_rowsum(a0, ones, s0);
        s1 = wmma_rowsum(a1, ones, s1);
        q0 = wmma_rowsum(b0, ones, q0);
        q1 = wmma_rowsum(b1, ones, q1);
    }

    // Tail: floats 3072..3135 = float2 indices 1536..1567, one per lane of
    // wave 0 (branch is wave-uniform, all 32 lanes active).
    if (tid < 32) {
        const float2* __restrict__ xp2 = (const float2*)xp4;
        float2 v = xp2[1536 + tid];
        v2f a; a.x = v.x;       a.y = v.y;
        v2f b; b.x = v.x * v.x; b.y = v.y * v.y;
        s0 = wmma_rowsum(a, ones, s0);
        q0 = wmma_rowsum(b, ones, q0);
    }

    // Fold C-matrices: per-lane sum of 8 VGPRs covers M=0..7 (lanes 0-15) or
    // M=8..15 (lanes 16-31); one xor-16 shuffle completes the wave total.
    v8f tS = s0 + s1;
    v8f tQ = q0 + q1;
    float s = tS[0] + tS[1] + tS[2] + tS[3] + tS[4] + tS[5] + tS[6] + tS[7];
    float q = tQ[0] + tQ[1] + tQ[2] + tQ[3] + tQ[4] + tQ[5] + tQ[6] + tQ[7];
    s += __shfl_xor(s, 16, 32);
    q += __shfl_xor(q, 16, 32);

    __shared__ float ls[8];
    __shared__ float lq[8];
    const int wave = tid >> 5;
    const int lane = tid & 31;
    if (lane == 0) { ls[wave] = s; lq[wave] = q; }
    __syncthreads();
    if (tid == 0) {
        float S = 0.f, Q = 0.f;
        for (int w = 0; w < 8; ++w) { S += ls[w]; Q += lq[w]; }
        psum[plane] = S;
        psq [plane] = Q;
    }
}

// ---------------------------------------------------------------------------
// Pass 2: fold 64 plane-partials per channel, produce scale/shift.
// ---------------------------------------------------------------------------
__global__ __launch_bounds__(C_DIM) void bn_finalize(const float* __restrict__ psum,
                                                     const float* __restrict__ psq,
                                                     const float* __restrict__ weight,
                                                     const float* __restrict__ bias,
                                                     float* __restrict__ scale,
                                                     float* __restrict__ shift) {
    const int c = threadIdx.x;
    float S = 0.f, Q = 0.f;
    for (int n = 0; n < N_DIM; ++n) {
        S += psum[n * C_DIM + c];
        Q += psq [n * C_DIM + c];
    }
    const float inv_n = 1.0f / (float)NUMEL_PER_CH;
    const float mean = S * inv_n;
    const float var  = Q * inv_n - mean * mean;
    const float istd = rsqrtf(var + BN_EPS);
    const float sc   = weight[c] * istd;
    scale[c] = sc;
    shift[c] = bias[c] - mean * sc;
}

// ---------------------------------------------------------------------------
// Pass 3: y = x*scale[c] + shift[c]. x (103 MB) should still be resident in
// the 192 MB L2 from pass 1; write y non-temporally so it doesn't evict x.
// ---------------------------------------------------------------------------
__global__ __launch_bounds__(256) void bn_apply(const float* __restrict__ x,
                                                const float* __restrict__ scale,
                                                const float* __restrict__ shift,
                                                float* __restrict__ y,
                                                int n4) {
    __shared__ float ssc[C_DIM];
    __shared__ float ssh[C_DIM];
    if (threadIdx.x < C_DIM) {
        ssc[threadIdx.x] = scale[threadIdx.x];
        ssh[threadIdx.x] = shift[threadIdx.x];
    }
    __syncthreads();

    const v4f* __restrict__ x4 = (const v4f*)x;
    v4f* __restrict__ y4 = (v4f*)y;
    const int stride = (int)gridDim.x * 256;

    for (int i = (int)blockIdx.x * 256 + (int)threadIdx.x; i < n4; i += stride) {
        if (i + stride < n4) {
            __builtin_prefetch(x4 + i + stride, 0, 1);   // global_prefetch_b8
        }
        const int c = (i / (HW / 4)) % C_DIM;            // 784 float4 per plane
        v4f v = x4[i];
        const float sc = ssc[c];
        const float sh = ssh[c];
        v4f r;
        r.x = v.x * sc + sh;
        r.y = v.y * sc + sh;
        r.z = v.z * sc + sh;
        r.w = v.w * sc + sh;
        __builtin_nontemporal_store(r, y4 + i);          // NT b128 store
    }
}

// ---------------------------------------------------------------------------
extern "C" void kernel_launch(void* const* d_in, const int* in_sizes, int n_in,
                              void* d_out, int out_size, void* d_ws, size_t ws_size,
                              hipStream_t stream) {
    const float* x = (const float*)d_in[0];
    const float* w = (const float*)d_in[1];
    const float* b = (const float*)d_in[2];
    float* y  = (float*)d_out;
    float* ws = (float*)d_ws;

    float* psum  = ws;                   // PLANES floats
    float* psq   = ws + PLANES;          // PLANES floats
    float* scale = ws + 2 * PLANES;      // C_DIM floats
    float* shift = scale + C_DIM;        // C_DIM floats

    dim3 g1(C_DIM, N_DIM);
    bn_partials<<<g1, 256, 0, stream>>>(x, psum, psq);
    bn_finalize<<<1, C_DIM, 0, stream>>>(psum, psq, w, b, scale, shift);

    const int n4 = out_size / 4;
    bn_apply<<<4096, 256, 0, stream>>>(x, scale, shift, y, n4);
}